// AAconv_19335942767139
// MI455X (gfx1250) — compile-verified
//
#include <hip/hip_runtime.h>
#include <hip/hip_bf16.h>

typedef float v2f __attribute__((ext_vector_type(2)));
typedef float v8f __attribute__((ext_vector_type(8)));

#define HW 1024

// ---------------------------------------------------------------------------
// Direct 3x3 conv, pad=1, NCHW / OIHW, B=16, H=W=32. One thread per output.
// ---------------------------------------------------------------------------
__global__ void conv3x3_kernel(const float* __restrict__ x,
                               const float* __restrict__ w,
                               const float* __restrict__ bias,
                               float* __restrict__ y,
                               int CI, int CO) {
    int idx = blockIdx.x * blockDim.x + threadIdx.x;
    int total = 16 * CO * HW;
    if (idx >= total) return;
    int px = idx & 31;
    int py = (idx >> 5) & 31;
    int oc = (idx >> 10) % CO;
    int b  = idx / (CO << 10);

    float acc = bias[oc];
    const float* wb = w + oc * CI * 9;
    const float* xb = x + b * CI * HW;
    for (int ic = 0; ic < CI; ++ic) {
        const float* xc = xb + ic * HW;
        const float* wc = wb + ic * 9;
#pragma unroll
        for (int ky = 0; ky < 3; ++ky) {
            int iy = py + ky - 1;
            if ((unsigned)iy >= 32u) continue;
#pragma unroll
            for (int kx = 0; kx < 3; ++kx) {
                int ix = px + kx - 1;
                if ((unsigned)ix >= 32u) continue;
                acc += wc[ky * 3 + kx] * xc[iy * 32 + ix];
            }
        }
    }
    y[idx] = acc;
}

// ---------------------------------------------------------------------------
// Attention: one workgroup = 4 waves (128 threads) per (batch, head,
// 16-row query strip), sharing one 64KB logits strip in LDS.
//   - QK^T via V_WMMA_F32_16X16X4_F32 (K = dkh = 4, exact fit);
//     wave w computes column tiles [w*16, w*16+16).
//   - relative-position logits added in a 128-thread strided pass.
//   - softmax + weighted-V (dvh = 1): wave w owns query rows [4w, 4w+4),
//     wave32 shuffle reductions.
// 4 waves per 64KB strip -> ~5 blocks (20 waves) resident per 320KB WGP,
// vs ~1 wave/SIMD with single-wave blocks: hides exp/LDS latency.
// qkv layout: (16, 36, 1024) = [q(16) | k(16) | v(4)], head channels n*4+d.
// ---------------------------------------------------------------------------
__global__ __launch_bounds__(128) void aac_attn_kernel(
        const float* __restrict__ qkv,
        const float* __restrict__ relw,   // (63,4)
        const float* __restrict__ relh,   // (63,4)
        float* __restrict__ attn) {       // (16, 4, 1024)
    __shared__ float lg[16 * HW];         // 64 KB logits strip

    const int tid   = threadIdx.x;
    const int lane  = tid & 31;
    const int wave  = tid >> 5;
    const int strip = blockIdx.x & 63;
    const int head  = (blockIdx.x >> 6) & 3;
    const int b     = blockIdx.x >> 8;
    const int i0    = strip * 16;

    const float* qb = qkv + (b * 36 + head * 4) * HW;
    const float* kb = qkv + (b * 36 + 16 + head * 4) * HW;
    const float* vb = qkv + (b * 36 + 32 + head) * HW;

    const int half = lane >> 4;   // 0: K=0,1  1: K=2,3 (32-bit A/B layout)
    const int l16  = lane & 15;

    // A operand: A[m][k] = 0.5 * q[k][i0+m]  (scale = dkh^-0.5 = 0.5)
    v2f a;
    a.x = 0.5f * qb[(2 * half + 0) * HW + i0 + l16];
    a.y = 0.5f * qb[(2 * half + 1) * HW + i0 + l16];

    for (int jt = wave * 16; jt < wave * 16 + 16; ++jt) {
        const int j0 = jt * 16;
        v2f bv;                   // B[k][n] = k[k][j0+n]
        bv.x = kb[(2 * half + 0) * HW + j0 + l16];
        bv.y = kb[(2 * half + 1) * HW + j0 + l16];
        v8f c = {};
        c = __builtin_amdgcn_wmma_f32_16x16x4_f32(
                false, a, false, bv, (short)0, c, false, false);
        // C/D layout: VGPR r -> row r (lanes 0-15) / row r+8 (lanes 16-31)
#pragma unroll
        for (int r = 0; r < 8; ++r)
            lg[(half * 8 + r) * HW + j0 + l16] = c[r];
    }
    __syncthreads();

    // Add relative logits: lg[m][j] += q_m . relh[xk-xq+31] + q_m . relw[yk-yq+31]
    // (row m uniform within each 32-lane group since 128 | 1024)
    for (int t0 = 0; t0 < 16 * HW; t0 += 128) {
        int t  = t0 + tid;
        int m  = t >> 10;
        int j  = t & (HW - 1);
        int iq = i0 + m, xq = iq >> 5, yq = iq & 31;
        int xk = j >> 5, yk = j & 31;
        float q0 = 0.5f * qb[0 * HW + iq];
        float q1 = 0.5f * qb[1 * HW + iq];
        float q2 = 0.5f * qb[2 * HW + iq];
        float q3 = 0.5f * qb[3 * HW + iq];
        const float* rh = relh + (xk - xq + 31) * 4;
        const float* rw = relw + (yk - yq + 31) * 4;
        lg[t] += q0 * rh[0] + q1 * rh[1] + q2 * rh[2] + q3 * rh[3]
               + q0 * rw[0] + q1 * rw[1] + q2 * rw[2] + q3 * rw[3];
    }
    __syncthreads();

    // Per-row softmax + weighted sum of v (dvh = 1); wave w owns 4 rows.
    for (int m = wave * 4; m < wave * 4 + 4; ++m) {
        float mx = -1e30f;
        for (int j = lane; j < HW; j += 32)
            mx = fmaxf(mx, lg[m * HW + j]);
#pragma unroll
        for (int s = 16; s; s >>= 1) mx = fmaxf(mx, __shfl_xor(mx, s, 32));

        float se = 0.f, sv = 0.f;
        for (int j = lane; j < HW; j += 32) {
            float e = __expf(lg[m * HW + j] - mx);
            se += e;
            sv += e * vb[j];
        }
#pragma unroll
        for (int s = 16; s; s >>= 1) {
            se += __shfl_xor(se, s, 32);
            sv += __shfl_xor(sv, s, 32);
        }
        if (lane == 0)
            attn[(b * 4 + head) * HW + i0 + m] = sv / se;
    }
}

// ---------------------------------------------------------------------------
// Concat conv_out with (1x1 conv of attn), optional ReLU.
// y: (16, CO, 1024); conv_out: (16, CO-4, 1024); attn: (16, 4, 1024)
// ---------------------------------------------------------------------------
__global__ void assemble_kernel(const float* __restrict__ conv_out,
                                const float* __restrict__ attn,
                                const float* __restrict__ attn_w,  // (4,4)
                                const float* __restrict__ attn_b,  // (4,)
                                float* __restrict__ y,
                                int CO, int do_relu) {
    int idx = blockIdx.x * blockDim.x + threadIdx.x;
    int total = 16 * CO * HW;
    if (idx >= total) return;
    int p = idx & (HW - 1);
    int c = (idx >> 10) % CO;
    int b = idx / (CO << 10);
    int cc = CO - 4;
    float v;
    if (c < cc) {
        v = conv_out[(b * cc + c) * HW + p];
    } else {
        int oc = c - cc;
        v = attn_b[oc];
#pragma unroll
        for (int ic = 0; ic < 4; ++ic)
            v += attn_w[oc * 4 + ic] * attn[(b * 4 + ic) * HW + p];
    }
    if (do_relu) v = fmaxf(v, 0.f);
    y[idx] = v;
}

// ---------------------------------------------------------------------------
// Final FC over 6 channels: out[b,p] = sum_c h2[b,c,p]*fc_w[c] + fc_b
// ---------------------------------------------------------------------------
__global__ void fc_kernel(const float* __restrict__ h2,
                          const float* __restrict__ fc_w,
                          const float* __restrict__ fc_b,
                          float* __restrict__ out) {
    int idx = blockIdx.x * blockDim.x + threadIdx.x;
    if (idx >= 16 * HW) return;
    int p = idx & (HW - 1);
    int b = idx >> 10;
    float acc = fc_b[0];
#pragma unroll
    for (int c = 0; c < 6; ++c)
        acc += h2[(b * 6 + c) * HW + p] * fc_w[c];
    out[idx] = acc;
}

// ---------------------------------------------------------------------------
extern "C" void kernel_launch(void* const* d_in, const int* in_sizes, int n_in,
                              void* d_out, int out_size, void* d_ws, size_t ws_size,
                              hipStream_t stream) {
    const float* x = (const float*)d_in[0];

    float* ws   = (float*)d_ws;
    float* h0   = ws;                        // 16*32*1024
    float* h1   = h0 + 16 * 32 * HW;         // 16*32*1024
    float* h2   = h1 + 16 * 32 * HW;         // 16*6*1024
    float* cbuf = h2 + 16 * 6 * HW;          // up to 16*28*1024
    float* qbuf = cbuf + 16 * 28 * HW;       // 16*36*1024
    float* abuf = qbuf + 16 * 36 * HW;       // 16*4*1024

    const int CI[3] = {8, 32, 32};
    const int CO[3] = {32, 32, 6};
    float* louts[3] = {h0, h1, h2};

    const float* lin = x;
    for (int l = 0; l < 3; ++l) {
        const float* conv_w = (const float*)d_in[1 + l * 8 + 0];
        const float* conv_b = (const float*)d_in[1 + l * 8 + 1];
        const float* qkv_w  = (const float*)d_in[1 + l * 8 + 2];
        const float* qkv_b  = (const float*)d_in[1 + l * 8 + 3];
        const float* attn_w = (const float*)d_in[1 + l * 8 + 4];
        const float* attn_b = (const float*)d_in[1 + l * 8 + 5];
        const float* relw   = (const float*)d_in[1 + l * 8 + 6];
        const float* relh   = (const float*)d_in[1 + l * 8 + 7];

        int coc = CO[l] - 4;
        int tot;

        tot = 16 * coc * HW;
        conv3x3_kernel<<<(tot + 255) / 256, 256, 0, stream>>>(
            lin, conv_w, conv_b, cbuf, CI[l], coc);

        tot = 16 * 36 * HW;
        conv3x3_kernel<<<(tot + 255) / 256, 256, 0, stream>>>(
            lin, qkv_w, qkv_b, qbuf, CI[l], 36);

        // 16 batches * 4 heads * 64 strips of 16 query rows; 4 waves/block
        aac_attn_kernel<<<16 * 4 * 64, 128, 0, stream>>>(qbuf, relw, relh, abuf);

        tot = 16 * CO[l] * HW;
        assemble_kernel<<<(tot + 255) / 256, 256, 0, stream>>>(
            cbuf, abuf, attn_w, attn_b, louts[l], CO[l], l < 2 ? 1 : 0);

        lin = louts[l];
    }

    const float* fc_w = (const float*)d_in[25];
    const float* fc_b = (const float*)d_in[26];
    fc_kernel<<<(16 * HW + 255) / 256, 256, 0, stream>>>(h2, fc_w, fc_b, (float*)d_out);
}